// ControlNodeWarp_2989297238219
// MI455X (gfx1250) — compile-verified
//
#include <hip/hip_runtime.h>
#include <hip/hip_bf16.h>
#include <math.h>

typedef float v2f __attribute__((ext_vector_type(2)));
typedef float v8f __attribute__((ext_vector_type(8)));

#define MAX_M 512
#define EPSW 1e-7f

// Branch-free sorted insert of (c, ci) into ascending triple (v0<=v1<=v2).
// Lowers to v_cndmask selects -> no EXEC divergence around the WMMA loop.
__device__ __forceinline__ void insert3(float c, int ci,
                                        float& v0, float& v1, float& v2,
                                        int& i0, int& i1, int& i2) {
    bool lt0 = c < v0, lt1 = c < v1, lt2 = c < v2;
    v2 = lt1 ? v1 : (lt2 ? c : v2);  i2 = lt1 ? i1 : (lt2 ? ci : i2);
    v1 = lt0 ? v0 : (lt1 ? c : v1);  i1 = lt0 ? i0 : (lt1 ? ci : i1);
    v0 = lt0 ? c  : v0;              i0 = lt0 ? ci : i0;
}

__global__ __launch_bounds__(256)
void cnw_kernel(const float* __restrict__ x,
                const float* __restrict__ nodes,
                const float* __restrict__ rraw,
                const float* __restrict__ wraw,
                const float* __restrict__ dxyz,
                const float* __restrict__ drot,
                const float* __restrict__ dscale,
                float* __restrict__ out, int N, int M) {
    // LDS node tables (M<=512): A-matrix rows {-2nx,-2ny,-2nz,|n|^2} (8 KB)
    // and deform table 3 x float4 per node:
    //   d0 = {inv2r2, sigw, dxyz0, dxyz1}
    //   d1 = {dxyz2, drot0, drot1, drot2}
    //   d2 = {drot3, dscale0, dscale1, dscale2}   (24 KB)
    __shared__ float  ldsA[MAX_M * 4];
    __shared__ float4 ldsD[MAX_M * 3];

    for (int m = threadIdx.x; m < M; m += blockDim.x) {
        float nx = nodes[m * 3 + 0];
        float ny = nodes[m * 3 + 1];
        float nz = nodes[m * 3 + 2];
        ldsA[m * 4 + 0] = -2.0f * nx;
        ldsA[m * 4 + 1] = -2.0f * ny;
        ldsA[m * 4 + 2] = -2.0f * nz;
        ldsA[m * 4 + 3] = nx * nx + ny * ny + nz * nz;
        // radius = exp(raw); 1/(2 r^2) = 0.5 * exp(-2*raw)
        float inv2r2 = 0.5f * __expf(-2.0f * rraw[m]);
        float sigw   = 1.0f / (1.0f + __expf(-wraw[m]));
        ldsD[m * 3 + 0] = make_float4(inv2r2, sigw, dxyz[m * 3 + 0], dxyz[m * 3 + 1]);
        ldsD[m * 3 + 1] = make_float4(dxyz[m * 3 + 2], drot[m * 4 + 0],
                                      drot[m * 4 + 1], drot[m * 4 + 2]);
        ldsD[m * 3 + 2] = make_float4(drot[m * 4 + 3], dscale[m * 3 + 0],
                                      dscale[m * 3 + 1], dscale[m * 3 + 2]);
    }
    __syncthreads();

    const int lane = threadIdx.x & 31;      // wave32
    const int hi   = lane >> 4;             // lane half: 0 -> K=0,1 / M=r ; 1 -> K=2,3 / M=r+8
    const int pl   = lane & 15;
    const int wave = threadIdx.x >> 5;      // 8 waves / block
    const int base = (blockIdx.x * 8 + wave) * 16;
    const int p    = base + pl;             // this lane-pair's point
    const int pc   = (p < N) ? p : (N - 1); // clamp loads; stores predicated

    // B matrix (4x16 f32): column = point, rows = {px,py,pz,1}.
    // Layout mirror of A: lanes 0-15 hold K=0 (B.x) and K=1 (B.y); lanes 16-31 hold K=2,K=3.
    float px = x[pc * 3 + 0];
    float py = x[pc * 3 + 1];
    float pz = x[pc * 3 + 2];
    float psq = px * px + py * py + pz * pz;
    v2f b;
    b.x = hi ? pz : px;
    b.y = hi ? 1.0f : py;

    // Running top-3 (smallest shifted-d2 = -2 n.p + |n|^2) for this lane's half.
    float v0 = INFINITY, v1 = INFINITY, v2c = INFINITY;
    int   i0 = 0, i1 = 0, i2 = 0;

    const int ntiles = M >> 4;
    const int acol   = hi * 2;

    // Software-pipelined A-tile loads: prefetch tile t+1 before consuming t so
    // the ds_load latency overlaps the ~75-VALU select chain (wait dscnt<=1,
    // not 0, ahead of each WMMA).
    v2f a;
    {
        int m = pl;
        a.x = ldsA[m * 4 + acol + 0];
        a.y = ldsA[m * 4 + acol + 1];
    }
    int t = 0;
    for (; t + 1 < ntiles; ++t) {
        v2f an;
        {
            int m = ((t + 1) << 4) + pl;
            an.x = ldsA[m * 4 + acol + 0];
            an.y = ldsA[m * 4 + acol + 1];
        }
        v8f c = {0.f, 0.f, 0.f, 0.f, 0.f, 0.f, 0.f, 0.f};
        // D[m_tile, point] = -2 n.p + |n|^2  (per-point |p|^2 added later; ordering invariant)
        v8f d = __builtin_amdgcn_wmma_f32_16x16x4_f32(
            /*neg_a=*/false, a, /*neg_b=*/false, b,
            /*c_mod=*/(short)0, c, /*reuse_a=*/false, /*reuse_b=*/false);
        int mbase = (t << 4) + hi * 8;
#pragma unroll
        for (int j = 0; j < 8; ++j)
            insert3(d[j], mbase + j, v0, v1, v2c, i0, i1, i2);
        a = an;
    }
    {   // peeled last tile (no prefetch, no in-loop tail branch)
        v8f c = {0.f, 0.f, 0.f, 0.f, 0.f, 0.f, 0.f, 0.f};
        v8f d = __builtin_amdgcn_wmma_f32_16x16x4_f32(
            false, a, false, b, (short)0, c, false, false);
        int mbase = (t << 4) + hi * 8;
#pragma unroll
        for (int j = 0; j < 8; ++j)
            insert3(d[j], mbase + j, v0, v1, v2c, i0, i1, i2);
    }

    // Merge the two half-lane triples (disjoint node sets) across the lane pair.
    float u0 = __shfl_xor(v0, 16, 32);  int j0 = __shfl_xor(i0, 16, 32);
    float u1 = __shfl_xor(v1, 16, 32);  int j1 = __shfl_xor(i1, 16, 32);
    float u2 = __shfl_xor(v2c, 16, 32); int j2 = __shfl_xor(i2, 16, 32);
    insert3(u0, j0, v0, v1, v2c, i0, i1, i2);
    insert3(u1, j1, v0, v1, v2c, i0, i1, i2);
    insert3(u2, j2, v0, v1, v2c, i0, i1, i2);

    if (hi == 0 && p < N) {
        int   idx[3] = {i0, i1, i2};
        float dsv[3] = {v0, v1, v2c};
        float wsum = 0.f;
        float tx = 0.f, ty = 0.f, tz = 0.f;
        float r0 = 0.f, r1 = 0.f, r2 = 0.f, r3 = 0.f;
        float s0 = 0.f, s1 = 0.f, s2 = 0.f;
#pragma unroll
        for (int k = 0; k < 3; ++k) {
            int m = idx[k];
            float4 da = ldsD[m * 3 + 0];
            float4 db = ldsD[m * 3 + 1];
            float4 dc = ldsD[m * 3 + 2];
            float d2 = fmaxf(dsv[k] + psq, 0.0f);           // restore |p|^2, clamp fp noise
            float w  = __expf(-d2 * da.x) * da.y + EPSW;    // gaussian * sigmoid + eps
            wsum += w;
            tx += w * da.z; ty += w * da.w; tz += w * db.x;
            r0 += w * db.y; r1 += w * db.z; r2 += w * db.w; r3 += w * dc.x;
            s0 += w * dc.y; s1 += w * dc.z; s2 += w * dc.w;
        }
        float inv = 1.0f / wsum;                 // weight normalization
        tx *= inv; ty *= inv; tz *= inv;
        s0 *= inv; s1 *= inv; s2 *= inv;
        // quaternion renorm is scale-invariant -> skip weight norm for it
        float rn = rsqrtf(r0 * r0 + r1 * r1 + r2 * r2 + r3 * r3);
        r0 *= rn; r1 *= rn; r2 *= rn; r3 *= rn;

        float* outT = out;
        float* outR = out + (size_t)3 * N;
        float* outS = out + (size_t)7 * N;
        outT[(size_t)p * 3 + 0] = tx;
        outT[(size_t)p * 3 + 1] = ty;
        outT[(size_t)p * 3 + 2] = tz;
        outR[(size_t)p * 4 + 0] = r0;
        outR[(size_t)p * 4 + 1] = r1;
        outR[(size_t)p * 4 + 2] = r2;
        outR[(size_t)p * 4 + 3] = r3;
        outS[(size_t)p * 3 + 0] = s0;
        outS[(size_t)p * 3 + 1] = s1;
        outS[(size_t)p * 3 + 2] = s2;
    }
}

extern "C" void kernel_launch(void* const* d_in, const int* in_sizes, int n_in,
                              void* d_out, int out_size, void* d_ws, size_t ws_size,
                              hipStream_t stream) {
    const float* x      = (const float*)d_in[0];
    const float* nodes  = (const float*)d_in[1];
    const float* rraw   = (const float*)d_in[2];
    const float* wraw   = (const float*)d_in[3];
    const float* dxyz   = (const float*)d_in[4];
    const float* drot   = (const float*)d_in[5];
    const float* dscale = (const float*)d_in[6];

    int N = in_sizes[0] / 3;
    int M = in_sizes[2];
    if (M > MAX_M) M = MAX_M;

    int blocks = (N + 127) / 128;   // 8 wave32s/block, 16 points/wave
    hipLaunchKernelGGL(cnw_kernel, dim3(blocks), dim3(256), 0, stream,
                       x, nodes, rraw, wraw, dxyz, drot, dscale,
                       (float*)d_out, N, M);
}